// BipartiteMatcher_36567351558159
// MI455X (gfx1250) — compile-verified
//
#include <hip/hip_runtime.h>
#include <stdint.h>

#define TPB 256
#define EPSF 1e-12f
#define K1_CHUNK 4
#define S_ROW 4     // row-splits for the column pass
#define S_COL 4     // column-splits for the row pass

typedef unsigned long long u64;

// ---- gfx1250 async-to-LDS helpers (inline asm, verified to assemble) -------
#define WAIT_ASYNC(n) asm volatile("s_wait_asynccnt %0" :: "i"(n) : "memory")

__device__ __forceinline__ void async_load_f4_to_lds(const float* gaddr, void* lds_dst) {
  unsigned lds_off = (unsigned)(unsigned long long)lds_dst;  // low 32 bits = LDS byte addr
  asm volatile("global_load_async_to_lds_b128 %0, %1, off"
               :: "v"(lds_off), "v"(gaddr) : "memory");
}

// ---- monotone float key + (value, first-index) packing for u64 atomicMax ---
__device__ __forceinline__ unsigned fkey(float f) {
  unsigned b = __float_as_uint(f);
  return ((int)b < 0) ? ~b : (b | 0x80000000u);
}
__device__ __forceinline__ float funkey(unsigned k) {
  unsigned b = ((int)k < 0) ? (k & 0x7FFFFFFFu) : ~k;
  return __uint_as_float(b);
}
// max value wins; on value tie, SMALLEST index wins (first occurrence).
__device__ __forceinline__ u64 packvi(float v, int idx) {
  return ((u64)fkey(v) << 32) | (unsigned)(~idx);
}
__device__ __forceinline__ int unpack_idx(u64 k) { return (int)~(unsigned)k; }

// ---------------- init: zero packed keys, counters, sentinels ----------------
__global__ void init_ws(u64* __restrict__ pk_col, u64* __restrict__ pk_row,
                        int* __restrict__ maskc, int* __restrict__ forced2,
                        int N, int M) {
  int i = blockIdx.x * TPB + threadIdx.x;
  if (i < M) pk_col[i] = 0ull;
  if (i < N) { pk_row[i] = 0ull; maskc[i] = 0; forced2[i] = -1; }
}

// ---------------- column pass (vectorized, M % 4 == 0) ----------------------
// Grid (colGroups, rowSplits). 4 columns/thread, async double-buffered LDS
// staging of K1_CHUNK rows, then one packed u64 atomicMax per column.
__global__ void __launch_bounds__(TPB)
col_pass_vec(const float* __restrict__ x, u64* __restrict__ pk_col,
             int N, int M, int rows_per) {
  __shared__ float buf[2][K1_CHUNK][TPB * 4];   // 32 KB
  const int t = threadIdx.x;
  const int ngroups = M >> 2;
  int g = blockIdx.x * TPB + t;
  const bool valid = (g < ngroups);
  if (!valid) g = ngroups - 1;                  // keep EXEC full
  const size_t col0 = (size_t)g * 4;

  const int j_begin = blockIdx.y * rows_per;
  const int j_end   = min(N, j_begin + rows_per);
  const int nrows   = j_end - j_begin;

  float bv0 = -INFINITY, bv1 = -INFINITY, bv2 = -INFINITY, bv3 = -INFINITY;
  int   bj0 = j_begin, bj1 = j_begin, bj2 = j_begin, bj3 = j_begin;

  const int nfull = nrows / K1_CHUNK;
  int p = 0;
  if (nfull > 0) {
#pragma unroll
    for (int r = 0; r < K1_CHUNK; ++r)
      async_load_f4_to_lds(x + (size_t)(j_begin + r) * M + col0, &buf[0][r][4 * t]);
  }
  for (int c = 0; c < nfull; ++c) {
    if (c + 1 < nfull) {
      const int jn = j_begin + (c + 1) * K1_CHUNK;
#pragma unroll
      for (int r = 0; r < K1_CHUNK; ++r)
        async_load_f4_to_lds(x + (size_t)(jn + r) * M + col0, &buf[p ^ 1][r][4 * t]);
      WAIT_ASYNC(K1_CHUNK);
    } else {
      WAIT_ASYNC(0);
    }
#pragma unroll
    for (int r = 0; r < K1_CHUNK; ++r) {
      const int j = j_begin + c * K1_CHUNK + r;
      const float4 v = *(const float4*)&buf[p][r][4 * t];
      if (v.x > bv0) { bv0 = v.x; bj0 = j; }
      if (v.y > bv1) { bv1 = v.y; bj1 = j; }
      if (v.z > bv2) { bv2 = v.z; bj2 = j; }
      if (v.w > bv3) { bv3 = v.w; bj3 = j; }
    }
    p ^= 1;
  }
  for (int j = j_begin + nfull * K1_CHUNK; j < j_end; ++j) {
    const float* rp = x + (size_t)j * M + col0;
    float v0 = rp[0], v1 = rp[1], v2 = rp[2], v3 = rp[3];
    if (v0 > bv0) { bv0 = v0; bj0 = j; }
    if (v1 > bv1) { bv1 = v1; bj1 = j; }
    if (v2 > bv2) { bv2 = v2; bj2 = j; }
    if (v3 > bv3) { bv3 = v3; bj3 = j; }
  }
  if (valid) {
    atomicMax(&pk_col[col0 + 0], packvi(bv0, bj0));
    atomicMax(&pk_col[col0 + 1], packvi(bv1, bj1));
    atomicMax(&pk_col[col0 + 2], packvi(bv2, bj2));
    atomicMax(&pk_col[col0 + 3], packvi(bv3, bj3));
  }
}

// Scalar fallback for general M (still row-split).
__global__ void __launch_bounds__(TPB)
col_pass_scalar(const float* __restrict__ x, u64* __restrict__ pk_col,
                int N, int M, int rows_per) {
  int m = blockIdx.x * TPB + threadIdx.x;
  if (m >= M) return;
  const int j_begin = blockIdx.y * rows_per;
  const int j_end   = min(N, j_begin + rows_per);
  float bv = -INFINITY; int bj = j_begin;
  for (int j = j_begin; j < j_end; ++j) {
    float v = x[(size_t)j * M + m];
    if (v > bv) { bv = v; bj = j; }
  }
  atomicMax(&pk_col[m], packvi(bv, bj));
}

// -------- decode packed column keys -> maxs/bti, init forcedA ---------------
__global__ void decode_col(const u64* __restrict__ pk_col, float* __restrict__ maxs,
                           int* __restrict__ bti, int* __restrict__ forcedA, int M) {
  int m = blockIdx.x * TPB + threadIdx.x;
  if (m < M) {
    u64 k = pk_col[m];
    maxs[m] = funkey((unsigned)(k >> 32));
    bti[m]  = unpack_idx(k);
    forcedA[m] = -1;
  }
}

// ---------------- row pass: grid (N, colSplits) -----------------------------
__global__ void __launch_bounds__(TPB)
row_pass(const float* __restrict__ x, const float* __restrict__ maxs,
         u64* __restrict__ pk_row, int* __restrict__ maskc,
         int N, int M, int elems_per) {
  const int j = blockIdx.x;
  const int t = threadIdx.x;
  const float* row = x + (size_t)j * M;
  const int m_begin = blockIdx.y * elems_per;
  const int m_end   = min(M, m_begin + elems_per);

  float bv = -INFINITY; int bi = 0x7fffffff; int cnt = 0;

  if (((M | m_begin) & 3) == 0) {               // 16B-aligned vector path
    const float4* row4 = (const float4*)row;
    const float4* max4 = (const float4*)maxs;
    const int g_begin = m_begin >> 2, g_end = m_end >> 2;
    for (int g = g_begin + t; g < g_end; g += TPB) {
      if (g + 4 * TPB < g_end) {
        __builtin_prefetch(&row4[g + 4 * TPB], 0, 1);   // -> global_prefetch
        __builtin_prefetch(&max4[g + 4 * TPB], 0, 1);
      }
      float4 v  = row4[g];
      float4 mx = max4[g];
      const int m = g * 4;
      if (v.x > bv) { bv = v.x; bi = m;     }
      if (v.y > bv) { bv = v.y; bi = m + 1; }
      if (v.z > bv) { bv = v.z; bi = m + 2; }
      if (v.w > bv) { bv = v.w; bi = m + 3; }
      cnt += (v.x + EPSF >= mx.x) + (v.y + EPSF >= mx.y) +
             (v.z + EPSF >= mx.z) + (v.w + EPSF >= mx.w);
    }
  } else {
    for (int m = m_begin + t; m < m_end; m += TPB) {
      float v = row[m];
      if (v > bv) { bv = v; bi = m; }
      cnt += (v + EPSF >= maxs[m]);
    }
  }

  __shared__ u64 sk[TPB]; __shared__ int sc[TPB];
  sk[t] = packvi(bv, bi); sc[t] = cnt;
  __syncthreads();
  for (int s = TPB / 2; s > 0; s >>= 1) {
    if (t < s) {
      u64 k2 = sk[t + s];
      if (k2 > sk[t]) sk[t] = k2;
      sc[t] += sc[t + s];
    }
    __syncthreads();
  }
  if (t == 0) {
    atomicMax(&pk_row[j], sk[0]);
    atomicAdd(&maskc[j], sc[0]);
  }
}

// ---- decode pargmax + last-write-wins scatter into forcedA -----------------
__global__ void scatter_gt(const u64* __restrict__ pk_row, int* __restrict__ pargmax,
                           int* __restrict__ forcedA, int N) {
  int j = blockIdx.x * TPB + threadIdx.x;
  if (j < N) {
    int pm = unpack_idx(pk_row[j]);
    pargmax[j] = pm;
    atomicMax(&forcedA[pm], j);
  }
}

// ---- match[m] + forced2[match[m]] = max m ----------------------------------
__global__ void match_scatter(const int* __restrict__ forcedA,
                              const int* __restrict__ bti,
                              int* __restrict__ forced2, int M) {
  int m = blockIdx.x * TPB + threadIdx.x;
  if (m < M) {
    int f  = forcedA[m];
    int mm = (f >= 0) ? f : bti[m];   // hit>0 <=> forcedA >= 0
    atomicMax(&forced2[mm], m);
  }
}

// ---- finalize new_match (int32) --------------------------------------------
__global__ void finalize(const int* __restrict__ forced2,
                         const int* __restrict__ maskc,
                         const int* __restrict__ pargmax,
                         int* __restrict__ out, int N) {
  int j = blockIdx.x * TPB + threadIdx.x;
  if (j < N) {
    int f2 = forced2[j];
    out[j] = (f2 >= 0) ? f2 : (maskc[j] > 0 ? pargmax[j] : -1);
  }
}

extern "C" void kernel_launch(void* const* d_in, const int* in_sizes, int n_in,
                              void* d_out, int out_size, void* d_ws, size_t ws_size,
                              hipStream_t stream) {
  const float* x = (const float*)d_in[0];
  const int N = out_size;                 // 512
  const int M = in_sizes[0] / N;          // 200000

  // workspace layout (u64 arrays first for alignment)
  u64*   pk_col  = (u64*)d_ws;            // M u64
  u64*   pk_row  = pk_col + M;            // N u64
  float* maxs    = (float*)(pk_row + N);  // M f32
  int*   bti     = (int*)(maxs + M);      // M i32
  int*   forcedA = bti + M;               // M i32
  int*   pargmax = forcedA + M;           // N i32
  int*   maskc   = pargmax + N;           // N i32
  int*   forced2 = maskc + N;             // N i32
  int*   out     = (int*)d_out;

  const int mn      = (M > N) ? M : N;
  const int blk_MN  = (mn + TPB - 1) / TPB;
  const int blk_M   = (M + TPB - 1) / TPB;
  const int blk_N   = (N + TPB - 1) / TPB;

  init_ws<<<blk_MN, TPB, 0, stream>>>(pk_col, pk_row, maskc, forced2, N, M);

  // column pass: split rows to fill the machine (784 blocks for 512x200000)
  int srow = (S_ROW < N) ? S_ROW : N;
  int rows_per = (N + srow - 1) / srow;
  int gy = (N + rows_per - 1) / rows_per;
  if ((M & 3) == 0) {
    int ngroups = M >> 2;
    dim3 grid((ngroups + TPB - 1) / TPB, gy);
    col_pass_vec<<<grid, TPB, 0, stream>>>(x, pk_col, N, M, rows_per);
  } else {
    dim3 grid(blk_M, gy);
    col_pass_scalar<<<grid, TPB, 0, stream>>>(x, pk_col, N, M, rows_per);
  }
  decode_col<<<blk_M, TPB, 0, stream>>>(pk_col, maxs, bti, forcedA, M);

  // row pass: split columns (2048 blocks for 512x200000)
  int elems_per = (M + S_COL - 1) / S_COL;
  elems_per = (elems_per + 3) & ~3;                 // keep 16B alignment per split
  int gy2 = (M + elems_per - 1) / elems_per;
  dim3 grid2((unsigned)N, (unsigned)gy2);
  row_pass<<<grid2, TPB, 0, stream>>>(x, maxs, pk_row, maskc, N, M, elems_per);

  scatter_gt<<<blk_N, TPB, 0, stream>>>(pk_row, pargmax, forcedA, N);
  match_scatter<<<blk_M, TPB, 0, stream>>>(forcedA, bti, forced2, M);
  finalize<<<blk_N, TPB, 0, stream>>>(forced2, maskc, pargmax, out, N);
}